// ModelSingle_2095944041052
// MI455X (gfx1250) — compile-verified
//
#include <hip/hip_runtime.h>

typedef __attribute__((ext_vector_type(2))) float v2f;
typedef __attribute__((ext_vector_type(8))) float v8f;

#define B_  32
#define S_  64
#define I_  333
#define O_  83
#define CT_ 666   // I_ * K (K=2) contraction length

// out[b,s,o] = sum_c P[b,c] * A_s[c,o] + bias[s,o]
//   P[b,2i]=x[b,s,i], P[b,2i+1]=x[b,s,i]^2
//   A_s row-major [CT_ x O_] at A + s*CT_*O_   (contiguous!)
//   bias[s,o] = sum_i A0[(s*I_+i)*O_ + o]
//
// Workgroup = (s, 16-col tile). 2 waves split the K range (332 + 334),
// each wave register-tiles BOTH 16-row batch tiles so the dominant
// B-matrix (A) stream is loaded once and feeds two WMMAs per K-step.
__global__ __launch_bounds__(64) void taylor_wmma_kernel(
    const float* __restrict__ x,   // [B,S,I]
    const float* __restrict__ A,   // [1,S,I,K,O]
    const float* __restrict__ A0,  // [1,S,I,O]
    float* __restrict__ out)       // [B,S,O]
{
    const int s    = blockIdx.x;
    const int nt   = blockIdx.y;         // 16-wide column tile
    const int tid  = threadIdx.x;
    const int wave = tid >> 5;           // K-split: 0 -> k<332, 1 -> k>=332
    const int lane = tid & 31;
    const int half = lane >> 4;          // K-half within WMMA A/B frags
    const int l16  = lane & 15;

    __shared__ float lds_part[64];
    __shared__ float lds_bias[16];
    __shared__ float redbuf[2 * 8 * 32]; // wave1 partial C tiles (2 KB)

    // ---- cooperative bias: 4 threads per column, 16 columns ----
    {
        const int colb = tid & 15;
        const int q    = tid >> 4;
        const int colc = min(nt * 16 + colb, O_ - 1);   // clamp: branch-free
        const float* a0p = A0 + (size_t)s * I_ * O_ + (size_t)q * O_ + colc;
        float p = 0.f;
        #pragma unroll 4
        for (int i = q; i < I_; i += 4) {
            p += *a0p;
            a0p += 4 * O_;
        }
        lds_part[tid] = p;
    }
    __syncthreads();
    if (tid < 16)
        lds_bias[tid] = lds_part[tid] + lds_part[tid + 16] +
                        lds_part[tid + 32] + lds_part[tid + 48];
    // (lds_bias consumed after the redbuf barrier below)

    // ---- GEMM: C[32x83] = P[32x666] @ A_s[666x83], K split across waves ----
    const int col     = nt * 16 + l16;
    const int colc    = min(col, O_ - 1);
    const int c_start = wave * 332;                     // wave0: 0, wave1: 332

    const float* xrow0 = x + ((size_t)l16        * S_ + s) * I_;  // M-tile 0
    const float* xrow1 = x + ((size_t)(l16 + 16) * S_ + s) * I_;  // M-tile 1
    const float* Ap    = A + (size_t)s * CT_ * O_;

    const float* xq0 = xrow0 + half + (c_start >> 1);
    const float* xq1 = xrow1 + half + (c_start >> 1);
    const float* bp  = Ap + (size_t)(c_start + 2 * half) * O_ + colc;

    v8f acc0 = {}, acc1 = {};

    // 83 constant steps per wave -> unroll by 4 + deep load pipelining.
    #pragma unroll 4
    for (int it = 0; it < 83; ++it) {
        const float x0 = xq0[0];
        const float x1 = xq1[0];
        v2f a0; a0.x = x0; a0.y = x0 * x0;              // {x, x^2}
        v2f a1; a1.x = x1; a1.y = x1 * x1;
        v2f b;  b.x = bp[0]; b.y = bp[O_];              // shared B-frag
        __builtin_prefetch(bp + 16 * O_, 0, 1);
        acc0 = __builtin_amdgcn_wmma_f32_16x16x4_f32(
            false, a0, false, b, (short)0, acc0, false, false);
        acc1 = __builtin_amdgcn_wmma_f32_16x16x4_f32(
            false, a1, false, b, (short)0, acc1, false, false);
        xq0 += 2; xq1 += 2; bp += 4 * O_;
    }

    if (wave == 1) {                                    // wave-uniform branch
        // peeled K-tail: k = 664,665 live only in half==0 lanes
        const float x0 = xrow0[I_ - 1];
        const float x1 = xrow1[I_ - 1];
        v2f a0; a0.x = x0; a0.y = x0 * x0;
        v2f a1; a1.x = x1; a1.y = x1 * x1;
        const int r0 = min(664 + 2 * half, CT_ - 1);    // clamp addr, zero frag
        const float b0 = Ap[(size_t)r0 * O_ + colc];
        const float b1 = Ap[(size_t)(CT_ - 1) * O_ + colc];
        v2f b;
        b.x = half ? 0.f : b0;
        b.y = half ? 0.f : b1;
        acc0 = __builtin_amdgcn_wmma_f32_16x16x4_f32(
            false, a0, false, b, (short)0, acc0, false, false);
        acc1 = __builtin_amdgcn_wmma_f32_16x16x4_f32(
            false, a1, false, b, (short)0, acc1, false, false);

        #pragma unroll
        for (int r = 0; r < 8; ++r) {
            redbuf[r * 32 + lane]       = acc0[r];
            redbuf[256 + r * 32 + lane] = acc1[r];
        }
    }
    __syncthreads();

    if (wave == 0) {
        const float bias = lds_bias[l16];
        if (col < O_) {
            #pragma unroll
            for (int r = 0; r < 8; ++r) {
                const int m = r + 8 * half;             // row within 16-row tile
                const float v0 = acc0[r] + redbuf[r * 32 + lane]       + bias;
                const float v1 = acc1[r] + redbuf[256 + r * 32 + lane] + bias;
                out[((size_t)m        * S_ + s) * O_ + col] = v0;   // b = m
                out[((size_t)(m + 16) * S_ + s) * O_ + col] = v1;   // b = m+16
            }
        }
    }
}

extern "C" void kernel_launch(void* const* d_in, const int* in_sizes, int n_in,
                              void* d_out, int out_size, void* d_ws, size_t ws_size,
                              hipStream_t stream) {
    const float* x  = (const float*)d_in[0];   // [32,64,333]
    const float* A  = (const float*)d_in[1];   // [1,64,333,2,83]
    const float* A0 = (const float*)d_in[2];   // [1,64,333,83]
    float* out = (float*)d_out;                // [32,64,83]

    dim3 grid(S_, (O_ + 15) / 16);             // (64, 6)
    taylor_wmma_kernel<<<grid, 64, 0, stream>>>(x, A, A0, out);
}